// RoIPool_76871324664411
// MI455X (gfx1250) — compile-verified
//
#include <hip/hip_runtime.h>
#include <cstdint>
#include <cstddef>

// RoIPool (mmcv/Caffe style) for MI455X / gfx1250.
// Strategy: per (ROI, 8-channel chunk), one TDM tensor_load_to_lds with
// iterate_enable DMAs eight 32x32 fp32 tiles (32KB) into LDS; double-buffered
// with s_wait_tensorcnt; 8 wave32 waves compute 8x49 bin maxima from LDS.

#define PHN 7
#define PWN 7
#define SWIN 32            // fixed crop window (matches reference S)
#define CB   8             // channels per chunk
#define HF   152
#define WF   152
#define CF   256
#define NTHREADS 256

typedef uint32_t u32x4 __attribute__((ext_vector_type(4)));
typedef int32_t  i32x4 __attribute__((ext_vector_type(4)));
typedef int32_t  i32x8 __attribute__((ext_vector_type(8)));

__global__ __launch_bounds__(NTHREADS) void roipool_tdm_kernel(
    const float* __restrict__ feat,
    const float* __restrict__ rois,
    const int*   __restrict__ stride_p,
    float* __restrict__ out)
{
    const int n   = blockIdx.x;
    const int tid = threadIdx.x;

    __shared__ float tile[2][CB][SWIN * SWIN];   // 2 x 32KB double buffer

    // ---- per-ROI params (uniform across block; every thread computes) ----
    const float scale = 1.0f / (float)stride_p[0];
    const float rb = rois[n * 5 + 0];
    const float x1 = rois[n * 5 + 1];
    const float y1 = rois[n * 5 + 2];
    const float x2 = rois[n * 5 + 3];
    const float y2 = rois[n * 5 + 4];
    const int b  = (int)rb;
    const int xs = (int)rintf(x1 * scale);   // v_rndne = round-half-even (jnp.round)
    const int ys = (int)rintf(y1 * scale);
    const int xe = (int)rintf(x2 * scale);
    const int ye = (int)rintf(y2 * scale);
    const int roih = max(ye - ys + 1, 1);
    const int roiw = max(xe - xs + 1, 1);
    const float binh = (float)roih / 7.0f;
    const float binw = (float)roiw / 7.0f;
    const int ah = min(max(ys, 0), HF - SWIN);
    const int aw = min(max(xs, 0), WF - SWIN);

    // ---- per-thread bin assignments: CB*49 = 392 outputs per chunk ----
    int a_cl[2], a_r0[2], a_r1[2], a_c0[2], a_c1[2], a_ob[2];
    bool a_v[2];
#pragma unroll
    for (int s = 0; s < 2; ++s) {
        const int idx = tid + s * NTHREADS;
        a_v[s] = (idx < CB * PHN * PWN);
        const int cl  = idx / 49;
        const int bin = idx - cl * 49;
        const int ph  = bin / 7;
        const int pw  = bin - ph * 7;
        // reference: hstart = clip(int(floor(ph*binh)) + ys, 0, H)  (float32 math)
        int hstart = min(max((int)floorf((float)ph * binh)        + ys, 0), HF);
        int hend   = min(max((int)ceilf ((float)(ph + 1) * binh)  + ys, 0), HF);
        int wstart = min(max((int)floorf((float)pw * binw)        + xs, 0), WF);
        int wend   = min(max((int)ceilf ((float)(pw + 1) * binw)  + xs, 0), WF);
        // intersect with the S x S crop window [ah, ah+32) x [aw, aw+32)
        a_r0[s] = max(hstart, ah) - ah;
        a_r1[s] = min(hend, ah + SWIN) - ah;
        a_c0[s] = max(wstart, aw) - aw;
        a_c1[s] = min(wend, aw + SWIN) - aw;
        a_cl[s] = cl;
        a_ob[s] = ph * 7 + pw;
    }

    // ---- TDM descriptor issue (wave 0 only; scalar/uniform values) ----
    const uint32_t plane = (uint32_t)(HF * WF);                     // 23104 elems
    const uint64_t gbase = (uint64_t)(uintptr_t)feat
        + 4ull * ((uint64_t)b * (uint64_t)(CF * HF * WF)
                  + (uint64_t)ah * WF + (uint64_t)aw);

    auto issue = [&](int chunk, int buf) {
        const uint64_t ga  = gbase + 4ull * (uint64_t)chunk * (uint64_t)CB * plane;
        const uint32_t lds = (uint32_t)(uintptr_t)(&tile[buf][0][0]);
        u32x4 g0; i32x8 g1; i32x4 g2; i32x4 g3;
        // ---- D# group 0 ----
        g0[0] = 1u;                                            // count=1, user mode
        g0[1] = lds;                                           // lds_addr (bytes)
        g0[2] = (uint32_t)ga;                                  // global_addr[31:0]
        g0[3] = (uint32_t)((ga >> 32) & 0x01FFFFFFu) | (2u << 30); // addr[56:32] | type=2
        // ---- D# group 1 ----
        g1[0] = (int)((2u << 16) | (1u << 19));                // data_size=4B, iterate_enable
        g1[1] = (int)((uint32_t)SWIN << 16);                   // tensor_dim0[15:0] @ bits 63:48
        g1[2] = (int)((uint32_t)SWIN << 16);                   // tensor_dim1[15:0] @ bits 95:80
        g1[3] = (int)((uint32_t)SWIN << 16);                   // tile_dim0 @ bits 127:112
        g1[4] = (int)SWIN;                                     // tile_dim1 @ bits 143:128 (tile_dim2=0)
        g1[5] = (int)WF;                                       // tensor_dim0_stride lo32
        g1[6] = (int)((plane & 0xFFFFu) << 16);                // dim0_stride hi16=0 | dim1_stride lo16
        g1[7] = (int)(plane >> 16);                            // dim1_stride hi32
        // ---- D# group 2 (iterate mode) ----
        g2[0] = 0;                                             // tensor_dim2
        g2[1] = (int)(SWIN * SWIN);                            // lds_addr_increment (elems)
        g2[2] = (int)plane;                                    // global_addr_increment lo32 (elems)
        g2[3] = (int)((uint32_t)(CB - 1) << 16);               // iterate_count @ bits 127:112
        // ---- D# group 3 ----
        g3[0] = 0; g3[1] = 0; g3[2] = 0; g3[3] = 0;
        // therock/clang-23 6-arg form: extra int32x8 operand before cpol
        i32x8 gx = {0, 0, 0, 0, 0, 0, 0, 0};
        __builtin_amdgcn_tensor_load_to_lds(g0, g1, g2, g3, gx, 0);
    };

    const int nchunks = CF / CB;                               // 32
    if (tid < 32) issue(0, 0);

    for (int k = 0; k < nchunks; ++k) {
        const int buf = k & 1;
        if (tid < 32) {
            if (k + 1 < nchunks) {
                issue(k + 1, buf ^ 1);                         // prefetch next chunk
                __builtin_amdgcn_s_wait_tensorcnt(1);          // chunk k landed (in-order)
            } else {
                __builtin_amdgcn_s_wait_tensorcnt(0);
            }
        }
        __syncthreads();

#pragma unroll
        for (int s = 0; s < 2; ++s) {
            if (!a_v[s]) continue;
            float m = -__builtin_inff();
            const float* t = &tile[buf][a_cl[s]][0];
            for (int r = a_r0[s]; r < a_r1[s]; ++r) {
                const float* row = t + r * SWIN;
                for (int c = a_c0[s]; c < a_c1[s]; ++c)
                    m = fmaxf(m, row[c]);
            }
            const float res = (m == -__builtin_inff()) ? 0.0f : m;  // empty bin -> 0
            const int ch = k * CB + a_cl[s];
            out[((size_t)n * CF + (size_t)ch) * (PHN * PWN) + a_ob[s]] = res;
        }
        __syncthreads();   // buf fully consumed before it is DMA-overwritten
    }
}

extern "C" void kernel_launch(void* const* d_in, const int* in_sizes, int n_in,
                              void* d_out, int out_size, void* d_ws, size_t ws_size,
                              hipStream_t stream) {
    (void)n_in; (void)d_ws; (void)ws_size; (void)out_size;
    const float* feat     = (const float*)d_in[0];
    const float* rois     = (const float*)d_in[1];
    const int*   stride_p = (const int*)d_in[2];
    float* out = (float*)d_out;
    const int N = in_sizes[1] / 5;   // 512 ROIs
    roipool_tdm_kernel<<<N, NTHREADS, 0, stream>>>(feat, rois, stride_p, out);
}